// SAGEGRU_10264971837847
// MI455X (gfx1250) — compile-verified
//
#include <hip/hip_runtime.h>

#define NN 20000
#define EE 320000
#define TT 64

typedef __attribute__((ext_vector_type(2))) float v2f;
typedef __attribute__((ext_vector_type(8))) float v8f;

__global__ void zero_f32(float* __restrict__ p, int n) {
  int i = blockIdx.x * blockDim.x + threadIdx.x;
  if (i < n) p[i] = 0.f;
}

__global__ void deg_kernel(const int* __restrict__ dst, float* __restrict__ deg, int E) {
  int e = blockIdx.x * blockDim.x + threadIdx.x;
  if (e < E) atomicAdd(&deg[dst[e]], 1.0f);
}

__global__ void invdeg_kernel(const float* __restrict__ deg, float* __restrict__ invdeg, int n) {
  int i = blockIdx.x * blockDim.x + threadIdx.x;
  if (i < n) { float d = deg[i]; invdeg[i] = (d > 0.f) ? (1.f / d) : 0.f; }
}

// scatter-add x[src]->agg[dst], F=16; 16 lanes per edge (perfectly coalesced per row)
__global__ void scatter16(const float* __restrict__ x, const int* __restrict__ src,
                          const int* __restrict__ dst, float* __restrict__ agg, int E) {
  int idx = blockIdx.x * blockDim.x + threadIdx.x;
  int e = idx >> 4, f = idx & 15;
  if (e < E) atomicAdd(&agg[(size_t)dst[e] * 16 + f], x[(size_t)src[e] * 16 + f]);
}

// scatter-add h[src]->agg[dst], F=128; one wave (32 lanes) per edge, 4 floats/lane
__global__ void scatter128(const float* __restrict__ h, const int* __restrict__ src,
                           const int* __restrict__ dst, float* __restrict__ agg, int E) {
  int idx = blockIdx.x * blockDim.x + threadIdx.x;
  int e = idx >> 5, lane = idx & 31;
  if (e < E) {
    const float* s = h + (size_t)src[e] * 128;
    float* d = agg + (size_t)dst[e] * 128;
#pragma unroll
    for (int k = 0; k < 128; k += 32) atomicAdd(&d[k + lane], s[k + lane]);
  }
}

// h1 = relu( (agg1*invdeg) @ W1l^T + b1l + xt @ W1r^T )   [N,16]x[16,128] -> [N,128]
__global__ __launch_bounds__(256) void sage1_wmma(
    const float* __restrict__ agg1, const float* __restrict__ xt,
    const float* __restrict__ invdeg,
    const float* __restrict__ W1l, const float* __restrict__ b1l,
    const float* __restrict__ W1r, float* __restrict__ h1) {
  __shared__ float sA[16 * 16];
  __shared__ float sX[16 * 16];
  int tid = threadIdx.x;
  int row0 = blockIdx.x * 16;
  {
    int r = tid >> 4, cc = tid & 15;
    sA[tid] = agg1[(size_t)(row0 + r) * 16 + cc] * invdeg[row0 + r];
    sX[tid] = xt[(size_t)(row0 + r) * 16 + cc];
  }
  __syncthreads();
  int wave = tid >> 5, lane = tid & 31, half = lane >> 4, l16 = lane & 15;
  int ncol = wave * 16 + l16;
  v8f c = {};
#pragma unroll
  for (int k = 0; k < 16; k += 4) {
    int kk = k + 2 * half;
    v2f aA = {sA[l16 * 16 + kk], sA[l16 * 16 + kk + 1]};
    v2f aX = {sX[l16 * 16 + kk], sX[l16 * 16 + kk + 1]};
    v2f bL = *(const v2f*)(W1l + (size_t)ncol * 16 + kk);
    v2f bR = *(const v2f*)(W1r + (size_t)ncol * 16 + kk);
    c = __builtin_amdgcn_wmma_f32_16x16x4_f32(false, aA, false, bL, (short)0, c, false, false);
    c = __builtin_amdgcn_wmma_f32_16x16x4_f32(false, aX, false, bR, (short)0, c, false, false);
  }
  float bias = b1l[ncol];
#pragma unroll
  for (int i = 0; i < 8; i++) {
    int m = i + 8 * half;
    h1[(size_t)(row0 + m) * 128 + ncol] = fmaxf(c[i] + bias, 0.f);
  }
}

// mean_over_nodes(relu((agg2*invdeg)@W2l^T + b2l + h1@W2r^T)) accumulated into Hg[t]
__global__ __launch_bounds__(256) void sage2_wmma_mean(
    const float* __restrict__ agg2, const float* __restrict__ h1,
    const float* __restrict__ invdeg,
    const float* __restrict__ W2l, const float* __restrict__ b2l,
    const float* __restrict__ W2r, float* __restrict__ Hgt, float invN) {
  __shared__ float sA[16 * 128];
  __shared__ float sH[16 * 128];
  int tid = threadIdx.x;
  int row0 = blockIdx.x * 16;
  for (int i = tid; i < 16 * 128; i += 256) {
    int r = i >> 7, cc = i & 127;
    sA[i] = agg2[(size_t)(row0 + r) * 128 + cc] * invdeg[row0 + r];
    sH[i] = h1[(size_t)(row0 + r) * 128 + cc];
  }
  __syncthreads();
  int wave = tid >> 5, lane = tid & 31, half = lane >> 4, l16 = lane & 15;
  int ncol = wave * 16 + l16;
  v8f c = {};
#pragma unroll 4
  for (int k = 0; k < 128; k += 4) {
    int kk = k + 2 * half;
    v2f aA = {sA[l16 * 128 + kk], sA[l16 * 128 + kk + 1]};
    v2f aH = {sH[l16 * 128 + kk], sH[l16 * 128 + kk + 1]};
    v2f bL = *(const v2f*)(W2l + (size_t)ncol * 128 + kk);
    v2f bR = *(const v2f*)(W2r + (size_t)ncol * 128 + kk);
    c = __builtin_amdgcn_wmma_f32_16x16x4_f32(false, aA, false, bL, (short)0, c, false, false);
    c = __builtin_amdgcn_wmma_f32_16x16x4_f32(false, aH, false, bR, (short)0, c, false, false);
  }
  float bias = b2l[ncol];
  float s = 0.f;
#pragma unroll
  for (int i = 0; i < 8; i++) s += fmaxf(c[i] + bias, 0.f);
  s += __shfl_xor(s, 16, 32);              // combine the two lane-halves (rows 0-7 + 8-15)
  if (half == 0) atomicAdd(&Hgt[ncol], s * invN);
}

// GI[t][g] = Hg[t] . W_ih[g] + b_ih[g]   (64 x 768, each a 128-dot)
__global__ void gi_kernel(const float* __restrict__ Hg, const float* __restrict__ W_ih,
                          const float* __restrict__ b_ih, float* __restrict__ GI) {
  int idx = blockIdx.x * blockDim.x + threadIdx.x;
  if (idx >= 64 * 768) return;
  int t = idx / 768, g = idx % 768;
  const float* x = Hg + t * 128;
  const float* w = W_ih + (size_t)g * 128;
  float acc = b_ih[g];
#pragma unroll 8
  for (int k = 0; k < 128; k++) acc += w[k] * x[k];
  GI[idx] = acc;
}

// sequential GRU over 64 steps (hidden 256) + MLP head; single 256-thread block
__global__ __launch_bounds__(256) void gru_head_kernel(
    const float* __restrict__ GI, const float* __restrict__ W_hh,
    const float* __restrict__ b_hh,
    const float* __restrict__ Wh1, const float* __restrict__ bh1,
    const float* __restrict__ Wh2, const float* __restrict__ bh2,
    float* __restrict__ out) {
  __shared__ float h[256];
  __shared__ float hid[64];
  int j = threadIdx.x;
  h[j] = 0.f;
  __syncthreads();
  float bhr = b_hh[j], bhz = b_hh[256 + j], bhn = b_hh[512 + j];
  const float* wr = W_hh + (size_t)j * 256;
  const float* wz = W_hh + (size_t)(256 + j) * 256;
  const float* wn = W_hh + (size_t)(512 + j) * 256;
  for (int t = 0; t < 64; t++) {
    float hr = bhr, hz = bhz, hn = bhn;
#pragma unroll 8
    for (int k = 0; k < 256; k++) {
      float hv = h[k];
      hr += wr[k] * hv;
      hz += wz[k] * hv;
      hn += wn[k] * hv;
    }
    float ir = GI[t * 768 + j], iz = GI[t * 768 + 256 + j], inn = GI[t * 768 + 512 + j];
    float r = 1.f / (1.f + expf(-(ir + hr)));
    float z = 1.f / (1.f + expf(-(iz + hz)));
    float n = tanhf(inn + r * hn);
    float hnew = (1.f - z) * n + z * h[j];
    __syncthreads();
    h[j] = hnew;
    __syncthreads();
  }
  if (j < 64) {
    const float* w = Wh1 + (size_t)j * 256;
    float acc = bh1[j];
    for (int k = 0; k < 256; k++) acc += w[k] * h[k];
    hid[j] = fmaxf(acc, 0.f);
  }
  __syncthreads();
  if (j == 0) {
    float acc = bh2[0];
    for (int k = 0; k < 64; k++) acc += Wh2[k] * hid[k];
    out[0] = acc;
  }
}

extern "C" void kernel_launch(void* const* d_in, const int* in_sizes, int n_in,
                              void* d_out, int out_size, void* d_ws, size_t ws_size,
                              hipStream_t stream) {
  const float* x_seq = (const float*)d_in[0];
  const int* edge = (const int*)d_in[1];
  const int* src = edge;
  const int* dst = edge + EE;
  const float* W1l = (const float*)d_in[2];
  const float* b1l = (const float*)d_in[3];
  const float* W1r = (const float*)d_in[4];
  const float* W2l = (const float*)d_in[5];
  const float* b2l = (const float*)d_in[6];
  const float* W2r = (const float*)d_in[7];
  const float* W_ih = (const float*)d_in[8];
  const float* W_hh = (const float*)d_in[9];
  const float* b_ih = (const float*)d_in[10];
  const float* b_hh = (const float*)d_in[11];
  const float* Wh1 = (const float*)d_in[12];
  const float* bh1 = (const float*)d_in[13];
  const float* Wh2 = (const float*)d_in[14];
  const float* bh2 = (const float*)d_in[15];

  float* ws = (float*)d_ws;
  float* deg = ws;                      // N
  float* invdeg = deg + NN;             // N
  float* agg1 = invdeg + NN;            // N*16
  float* h1 = agg1 + (size_t)NN * 16;   // N*128
  float* agg2 = h1 + (size_t)NN * 128;  // N*128
  float* Hg = agg2 + (size_t)NN * 128;  // T*128
  float* GI = Hg + TT * 128;            // T*768

  auto cdiv = [](long a, long b) { return (int)((a + b - 1) / b); };

  zero_f32<<<cdiv(NN, 256), 256, 0, stream>>>(deg, NN);
  zero_f32<<<cdiv(TT * 128, 256), 256, 0, stream>>>(Hg, TT * 128);
  deg_kernel<<<cdiv(EE, 256), 256, 0, stream>>>(dst, deg, EE);
  invdeg_kernel<<<cdiv(NN, 256), 256, 0, stream>>>(deg, invdeg, NN);

  const float invN = 1.0f / (float)NN;
  for (int t = 0; t < TT; t++) {
    const float* xt = x_seq + (size_t)t * NN * 16;
    zero_f32<<<cdiv((long)NN * 16, 256), 256, 0, stream>>>(agg1, NN * 16);
    scatter16<<<cdiv((long)EE * 16, 256), 256, 0, stream>>>(xt, src, dst, agg1, EE);
    sage1_wmma<<<NN / 16, 256, 0, stream>>>(agg1, xt, invdeg, W1l, b1l, W1r, h1);
    zero_f32<<<cdiv((long)NN * 128, 256), 256, 0, stream>>>(agg2, NN * 128);
    scatter128<<<cdiv((long)EE * 32, 256), 256, 0, stream>>>(h1, src, dst, agg2, EE);
    sage2_wmma_mean<<<NN / 16, 256, 0, stream>>>(agg2, h1, invdeg, W2l, b2l, W2r,
                                                 Hg + t * 128, invN);
  }
  gi_kernel<<<cdiv(64 * 768, 256), 256, 0, stream>>>(Hg, W_ih, b_ih, GI);
  gru_head_kernel<<<1, 256, 0, stream>>>(GI, W_hh, b_hh, Wh1, bh1, Wh2, bh2, (float*)d_out);
}